// ChunkedDotProdAttention_50886772523163
// MI455X (gfx1250) — compile-verified
//
#include <hip/hip_runtime.h>

typedef __attribute__((ext_vector_type(2))) float v2f;
typedef __attribute__((ext_vector_type(8))) float v8f;

#define D_MODEL 2048
#define NHEADS  16
#define DH      128          // D_MODEL / NHEADS
#define NCHUNK  32           // DH / 4 (fp32 WMMA K=4 chunks)
#define MASK_VAL (-50000.0f)
#define SCALE 0.08838834764831845f   // dh^-0.5 = 1/sqrt(128)

// lane <-> lane^16 exchange (group-of-32 swizzle, xor=0x10, and=0x1f)
static __device__ __forceinline__ float swap16(float x) {
  return __int_as_float(__builtin_amdgcn_ds_swizzle(__float_as_int(x), 0x401f));
}
// broadcast value held by lane `row` (0..15) to this lane
static __device__ __forceinline__ float row_bcast(float x, int row) {
  return __int_as_float(__builtin_amdgcn_ds_bpermute(row << 2, __float_as_int(x)));
}

__global__ void __launch_bounds__(128, 1)
attn_fwd_f32wmma(const float* __restrict__ Q, const float* __restrict__ K,
                 const float* __restrict__ V, float* __restrict__ Out,
                 int b, int n)
{
  const int lane = threadIdx.x & 31;
  const int li   = lane & 15;          // column index within 16
  const int lh   = lane >> 4;          // half-wave select
  const int ntiles = n >> 4;
  const int wt = (int)blockIdx.x * 4 + ((int)threadIdx.x >> 5);
  if (wt >= b * NHEADS * ntiles) return;   // wave-uniform guard

  const int it = wt % ntiles;          // 16-row query tile index
  const int bh = wt / ntiles;
  const int hh = bh % NHEADS;
  const int bb = bh / NHEADS;

  const long rs   = D_MODEL;                                  // row stride (floats)
  const long base = ((long)bb * n) * D_MODEL + (long)hh * DH; // head slice

  // ---- Q tile (16 x 128) in WMMA A/B lane layout, pre-scaled by 1/sqrt(dh).
  // lane l holds tile[l%16][4c + 2*(l/16) + {0,1}] as float2 -> matches both
  // the 16x4 fp32 A layout and the 4x16 B layout.
  v2f qa[NCHUNK];
  {
    const float* qp = Q + base + (long)(it * 16 + li) * rs + 2 * lh;
    #pragma unroll
    for (int c = 0; c < NCHUNK; ++c) {
      v2f t = *(const v2f*)(qp + 4 * c);
      qa[c].x = t.x * SCALE;
      qa[c].y = t.y * SCALE;
    }
  }

  v8f acc[8];                                  // O tile 16 x 128, D layout per 16-col slab
  #pragma unroll
  for (int d = 0; d < 8; ++d) acc[d] = (v8f){0.f,0.f,0.f,0.f,0.f,0.f,0.f,0.f};
  float m = -3.0e38f;                          // running row max (row i = li)
  float lsum = 0.0f;                           // running row sum

  for (int jt = 0; jt <= it; ++jt) {
    // ---- S^T (16j x 16i) = K_tile(16j x dh) * Q^T(dh x 16i), fp32 WMMA
    const float* kp = K + base + (long)(jt * 16 + li) * rs + 2 * lh;
    v8f st = (v8f){0.f,0.f,0.f,0.f,0.f,0.f,0.f,0.f};
    #pragma unroll
    for (int c = 0; c < NCHUNK; ++c) {
      v2f ka = *(const v2f*)(kp + 4 * c);
      st = __builtin_amdgcn_wmma_f32_16x16x4_f32(false, ka, false, qa[c],
                                                 (short)0, st, false, false);
    }

    float s[8];
    #pragma unroll
    for (int r = 0; r < 8; ++r) s[r] = st[r];

    if (jt == it) {   // causal mask inside diagonal tile: local j > local i
      #pragma unroll
      for (int r = 0; r < 8; ++r)
        s[r] = ((r + 8 * lh) > li) ? MASK_VAL : s[r];   // cndmask, EXEC untouched
    }

    // ---- online softmax; S^T reg r at this lane is (j = r+8*lh, i = li),
    // so per-row(i) stats are lane-local + one SWAPX16 combine.
    float tmax = s[0];
    #pragma unroll
    for (int r = 1; r < 8; ++r) tmax = fmaxf(tmax, s[r]);
    tmax = fmaxf(tmax, swap16(tmax));
    const float mn = fmaxf(m, tmax);
    const float alpha = __expf(m - mn);
    m = mn;

    float p[8], ts = 0.f;
    #pragma unroll
    for (int r = 0; r < 8; ++r) { p[r] = __expf(s[r] - mn); ts += p[r]; }
    ts += swap16(ts);
    lsum = lsum * alpha + ts;

    // ---- rescale O accumulators; acc[d] reg r holds row i = r + 8*lh
    float ar[8];
    #pragma unroll
    for (int r = 0; r < 8; ++r) ar[r] = row_bcast(alpha, r + 8 * lh);
    #pragma unroll
    for (int d = 0; d < 8; ++d) {
      #pragma unroll
      for (int r = 0; r < 8; ++r) acc[d][r] *= ar[r];
    }

    // ---- P^T (D layout) -> P (A layout): lane<->lane+16 swap + select
    float sp[8];
    #pragma unroll
    for (int r = 0; r < 8; ++r) sp[r] = swap16(p[r]);
    v2f pa[4];  // A chunks over j: jc covers j = 4jc + 2*lh + {0,1}
    pa[0].x = lh ? sp[2] : p[0];  pa[0].y = lh ? sp[3] : p[1];
    pa[1].x = lh ? sp[6] : p[4];  pa[1].y = lh ? sp[7] : p[5];
    pa[2].x = lh ? p[2]  : sp[0]; pa[2].y = lh ? p[3]  : sp[1];
    pa[3].x = lh ? p[6]  : sp[4]; pa[3].y = lh ? p[7]  : sp[5];

    // ---- O(16i x 128d) += P(16i x 16j) * V_tile(16j x 128d)
    // B operand: lane l holds V[j0 + 2*lh + {0,1}][d0 + l%16]
    const float* vp = V + base + (long)(jt * 16 + 2 * lh) * rs + li;
    #pragma unroll
    for (int dt = 0; dt < 8; ++dt) {
      #pragma unroll
      for (int jc = 0; jc < 4; ++jc) {
        const float* vb = vp + (long)(4 * jc) * rs + 16 * dt;
        v2f vB; vB.x = vb[0]; vB.y = vb[rs];
        acc[dt] = __builtin_amdgcn_wmma_f32_16x16x4_f32(false, pa[jc], false, vB,
                                                        (short)0, acc[dt], false, false);
      }
    }
  }

  // ---- epilogue: divide by row sums, store fp32
  const float linv = 1.0f / lsum;
  float lr[8];
  #pragma unroll
  for (int r = 0; r < 8; ++r) lr[r] = row_bcast(linv, r + 8 * lh);
  float* op = Out + base + (long)(it * 16) * rs;
  #pragma unroll
  for (int dt = 0; dt < 8; ++dt) {
    #pragma unroll
    for (int r = 0; r < 8; ++r)
      op[(long)(r + 8 * lh) * rs + 16 * dt + li] = acc[dt][r] * lr[r];
  }
}

extern "C" void kernel_launch(void* const* d_in, const int* in_sizes, int n_in,
                              void* d_out, int out_size, void* d_ws, size_t ws_size,
                              hipStream_t stream) {
  const float* q = (const float*)d_in[0];
  const float* k = (const float*)d_in[1];
  const float* v = (const float*)d_in[2];
  float* out = (float*)d_out;

  const int n = 2048;                       // sequence length (reference module)
  const int b = in_sizes[0] / (n * D_MODEL);
  const int waves  = b * NHEADS * (n >> 4); // one wave per 16-row query tile
  const int blocks = (waves + 3) / 4;       // 4 waves (128 threads) per block
  attn_fwd_f32wmma<<<blocks, 128, 0, stream>>>(q, k, v, out, b, n);
}